// FFM_21912923144767
// MI455X (gfx1250) — compile-verified
//
#include <hip/hip_runtime.h>
#include <math.h>

typedef __attribute__((ext_vector_type(2))) float v2f;
typedef __attribute__((ext_vector_type(8))) float v8f;

#define DENSE   13
#define SPARSE  26
#define EMB     8
#define FEAT    221            // 13 + 26*8
#define FIELD   39
#define VOCAB   100000
#define KPAD    224            // K padded to multiple of 4
#define TSTRIDE 228            // tile row stride (floats): mult of 4; 228*k mod 64 -> 16 distinct banks
#define WAVES   2
#define ROWS_PER_WAVE  16
#define ROWS_PER_BLOCK (WAVES * ROWS_PER_WAVE)   // 32

__global__ __launch_bounds__(WAVES * 32)
void ffm_fused_kernel(const float* __restrict__ x_dense,
                      const int*   __restrict__ x_sparse,
                      const float* __restrict__ emb_tables,
                      const float* __restrict__ field_matrix,
                      const float* __restrict__ w,
                      const float* __restrict__ bias,
                      float* __restrict__ out,
                      int Brows)
{
    // B operand: sV[k][n], n=0..7 -> V columns, n=8 -> w (linear term), n=9..15 -> 0
    __shared__ __align__(16) float sV[KPAD * 16];                     // 14336 B
    __shared__ __align__(16) float sT[WAVES][ROWS_PER_WAVE * TSTRIDE];// 2*14592 B

    const int tid  = threadIdx.x;
    const int wave = tid >> 5;
    const int lane = tid & 31;
    const int half = lane >> 4;   // 0: K pair {0,1}; 1: K pair {2,3}
    const int lrow = lane & 15;

    // -------- build extended B operand once per block --------
    // Tile column remap: k<208 -> embedding feature i = 13+k ; 208<=k<221 -> dense i = k-208
    for (int k = tid; k < KPAD; k += blockDim.x) {
        float col8 = 0.0f;
        if (k < FEAT) {
            int i = (k < SPARSE * EMB) ? (DENSE + k) : (k - SPARSE * EMB);
            int f = (i < DENSE) ? i : (((i - DENSE) >> 3) + DENSE);
            const float* src = field_matrix + ((size_t)i * FIELD + f) * EMB;
            #pragma unroll
            for (int e = 0; e < EMB; ++e) sV[k * 16 + e] = src[e];
            col8 = w[i];
        } else {
            #pragma unroll
            for (int e = 0; e < EMB; ++e) sV[k * 16 + e] = 0.0f;
        }
        sV[k * 16 + 8] = col8;
        #pragma unroll
        for (int e = 9; e < 16; ++e) sV[k * 16 + e] = 0.0f;
    }

    // -------- gather this wave's 16-row A tile into LDS --------
    const int rowBase = blockIdx.x * ROWS_PER_BLOCK + wave * ROWS_PER_WAVE;
    float* tile = sT[wave];

    // embeddings -> cols [0,208): 16 rows x 26 tables, 32 B each (two float4)
    for (int t = lane; t < ROWS_PER_WAVE * SPARSE; t += 32) {
        int r = t / SPARSE, j = t % SPARSE;
        int rg = rowBase + r; if (rg >= Brows) rg = Brows - 1;
        int idx = x_sparse[(size_t)rg * SPARSE + j];
        const float4* src = (const float4*)(emb_tables + ((size_t)j * VOCAB + idx) * EMB);
        float4 lo = src[0], hi = src[1];
        float4* dst = (float4*)&tile[r * TSTRIDE + j * EMB];
        dst[0] = lo; dst[1] = hi;
    }
    // dense -> cols [208,221)
    for (int t = lane; t < ROWS_PER_WAVE * DENSE; t += 32) {
        int r = t / DENSE, d = t % DENSE;
        int rg = rowBase + r; if (rg >= Brows) rg = Brows - 1;
        tile[r * TSTRIDE + SPARSE * EMB + d] = x_dense[(size_t)rg * DENSE + d];
    }
    // zero pad cols [221,224)
    for (int t = lane; t < ROWS_PER_WAVE * (KPAD - FEAT); t += 32) {
        int r = t / (KPAD - FEAT), c = t % (KPAD - FEAT);
        tile[r * TSTRIDE + FEAT + c] = 0.0f;
    }
    __syncthreads();

    // -------- 16x16 tile GEMM on the matrix pipe: K = 224 in steps of 4 --------
    // A frag (16x4 f32): lane half selects K pair; B frag mirrors with lane = N.
    v8f acc = {};
    #pragma unroll 4
    for (int kk = 0; kk < KPAD / 4; ++kk) {
        const int k0 = kk * 4 + half * 2;
        v2f a, bf;
        a.x  = tile[lrow * TSTRIDE + k0];
        a.y  = tile[lrow * TSTRIDE + k0 + 1];
        bf.x = sV[(k0    ) * 16 + lrow];
        bf.y = sV[(k0 + 1) * 16 + lrow];
        acc = __builtin_amdgcn_wmma_f32_16x16x4_f32(false, a, false, bf,
                                                    (short)0, acc, false, false);
    }

    // -------- epilogue: stage D through LDS (same-wave DS ops are in-order) --------
    float* sD = tile;   // reuse wave-private tile region; stride 17 avoids bank conflicts
    #pragma unroll
    for (int g = 0; g < 8; ++g)
        sD[(g + half * 8) * 17 + lrow] = acc[g];   // D[M][N], M = g + 8*half, N = lrow

    if (lane < 16) {
        int rg = rowBase + lrow;
        if (rg < Brows) {
            float lin = sD[lrow * 17 + 8] + bias[0];   // x.w + b
            float inter = 0.0f;
            #pragma unroll
            for (int n = 0; n < 8; ++n) { float v = sD[lrow * 17 + n]; inter += v * v; }
            float z = lin + inter;
            out[rg] = 1.0f / (1.0f + __expf(-z));
        }
    }
}

extern "C" void kernel_launch(void* const* d_in, const int* in_sizes, int n_in,
                              void* d_out, int out_size, void* d_ws, size_t ws_size,
                              hipStream_t stream) {
    const float* x_dense      = (const float*)d_in[0];
    const int*   x_sparse     = (const int*)  d_in[1];
    const float* emb_tables   = (const float*)d_in[2];
    const float* field_matrix = (const float*)d_in[3];
    const float* w            = (const float*)d_in[4];
    const float* b            = (const float*)d_in[5];
    float* out = (float*)d_out;

    const int Brows = in_sizes[0] / DENSE;
    const int nblk  = (Brows + ROWS_PER_BLOCK - 1) / ROWS_PER_BLOCK;
    ffm_fused_kernel<<<nblk, WAVES * 32, 0, stream>>>(
        x_dense, x_sparse, emb_tables, field_matrix, w, b, out, Brows);
}